// DynamicLookup_19043884990872
// MI455X (gfx1250) — compile-verified
//
#include <hip/hip_runtime.h>
#include <stdint.h>

// ---------------- configuration ----------------
#define EMPTY_KEY (-2147483647 - 1)   // INT_MIN sentinel (never a real key here)
#define IMAX      2147483647
#define VHS  4096     // vocab hash slots (pow2, load factor 0.5 for V=2048)
#define UHS  8192     // unique-set hash slots (pow2)
#define MAXU 4096     // bitonic sort capacity for uniques (pow2)
#define TMAX 2560     // >= T (2457)
#define PADK (-1)

__device__ __forceinline__ uint32_t ihash(int x) {
  return ((uint32_t)x) * 2654435761u;   // Knuth multiplicative hash
}

// ---------------- K0: init workspace hash tables ----------------
__global__ void k_init(int2* __restrict__ vslots, int* __restrict__ ukeys) {
  int i = blockIdx.x * blockDim.x + threadIdx.x;
  if (i < VHS) vslots[i] = make_int2(EMPTY_KEY, IMAX);
  if (i < UHS) ukeys[i] = EMPTY_KEY;
}

// ---------------- K1: build vocab hash (value -> first index) ----------------
__global__ void k_vocab_insert(const int* __restrict__ vocab, int V,
                               int2* __restrict__ vslots) {
  int i = blockIdx.x * blockDim.x + threadIdx.x;
  if (i >= V) return;
  int k = vocab[i];
  uint32_t h = ihash(k) & (VHS - 1);
  for (int p = 0; p < VHS; ++p) {
    int cur = atomicCAS(&vslots[h].x, EMPTY_KEY, k);
    if (cur == EMPTY_KEY || cur == k) {
      atomicMin(&vslots[h].y, i);   // first-match semantics for duplicate values
      return;
    }
    h = (h + 1) & (VHS - 1);
  }
}

// ---------------- K2: lookup + unique-set insert ----------------
__global__ __launch_bounds__(256)
void k_lookup(const int* __restrict__ flat, int N,
              const int2* __restrict__ vslots,
              int* __restrict__ ukeys,
              float* __restrict__ out, int V) {
  __shared__ int2 ls[VHS];   // 32 KB LDS copy of vocab hash table

  // CDNA5 async global->LDS copy (tracked by ASYNCcnt)
  for (int i = threadIdx.x; i < VHS; i += blockDim.x) {
    uint32_t lo = (uint32_t)(uintptr_t)(&ls[i]);
    uint64_t ga = (uint64_t)(uintptr_t)(&vslots[i]);
    asm volatile("global_load_async_to_lds_b64 %0, %1, off"
                 :: "v"(lo), "v"(ga) : "memory");
  }
  asm volatile("s_wait_asynccnt 0" ::: "memory");
  __syncthreads();

  int stride = gridDim.x * blockDim.x;
  for (int i = blockIdx.x * blockDim.x + threadIdx.x; i < N; i += stride) {
    if (i + stride < N) __builtin_prefetch(&flat[i + stride], 0, 1);
    int x = flat[i];

    // probe LDS hash table (ds_load_b64 per probe)
    uint32_t h = ihash(x) & (VHS - 1);
    int idx = V;
    for (int p = 0; p < VHS; ++p) {
      int2 s = ls[h];
      if (s.x == x) { idx = s.y; break; }
      if (s.x == EMPTY_KEY) break;
      h = (h + 1) & (VHS - 1);
    }
    out[i] = (float)idx;

    // insert into global unique set (dedup via atomicCAS)
    uint32_t u = ihash(x) & (UHS - 1);
    for (int p = 0; p < UHS; ++p) {
      int cur = ukeys[u];
      if (cur == x) break;
      if (cur == EMPTY_KEY) {
        int old = atomicCAS(&ukeys[u], EMPTY_KEY, x);
        if (old == EMPTY_KEY || old == x) break;
        if (old == x) break;
      }
      u = (u + 1) & (UHS - 1);
    }
  }
}

// ---------------- K3: table update (one workgroup) ----------------
__global__ __launch_bounds__(1024)
void k_finalize(const int* __restrict__ ukeys,
                const int* __restrict__ tkeys,
                const int* __restrict__ tvals,
                int T, float* __restrict__ outK, float* __restrict__ outV) {
  __shared__ int uq[MAXU];            // 16 KB
  __shared__ int sKeys[TMAX];         // 10 KB
  __shared__ int sVals[TMAX];         // 10 KB
  __shared__ int rnk[TMAX];           // 10 KB
  __shared__ unsigned char newm[MAXU];// 4 KB
  __shared__ int cntU, nNew, nExist;

  const int tid = threadIdx.x, bd = blockDim.x;
  if (tid == 0) { cntU = 0; nNew = 0; nExist = 0; }
  __syncthreads();

  // extract uniques from hash set; load table into LDS; count valid keys
  for (int i = tid; i < UHS; i += bd) {
    int k = ukeys[i];
    if (k != EMPTY_KEY) {
      int pos = atomicAdd(&cntU, 1);
      if (pos < MAXU) uq[pos] = k;
    }
  }
  int nEx = 0;
  for (int i = tid; i < T; i += bd) {
    int k = tkeys[i];
    sKeys[i] = k;
    sVals[i] = tvals[i];
    if (k != PADK) nEx++;
  }
  if (nEx) atomicAdd(&nExist, nEx);
  __syncthreads();

  int nU = cntU < MAXU ? cntU : MAXU;
  for (int i = tid; i < MAXU; i += bd) if (i >= nU) uq[i] = IMAX;
  __syncthreads();

  // bitonic ascending sort of uq[0..MAXU)
  for (int k2 = 2; k2 <= MAXU; k2 <<= 1) {
    for (int j = k2 >> 1; j > 0; j >>= 1) {
      for (int i = tid; i < MAXU; i += bd) {
        int ixj = i ^ j;
        if (ixj > i) {
          int a = uq[i], b = uq[ixj];
          bool up = ((i & k2) == 0);
          if ((a > b) == up) { uq[i] = b; uq[ixj] = a; }
        }
      }
      __syncthreads();
    }
  }

  const int nValidU = nU < T ? nU : T;   // jnp.unique(size=T) truncation

  // new_mask per unique (set difference uniq \ table_keys)
  int myNew = 0;
  for (int u = tid; u < nValidU; u += bd) {
    int x = uq[u];
    bool inTab = false;
    for (int j = 0; j < T; ++j)
      if (sKeys[j] == x && sKeys[j] != PADK) { inTab = true; break; }
    newm[u] = inTab ? 0 : 1;
    myNew += inTab ? 0 : 1;
  }
  if (myNew) atomicAdd(&nNew, myNew);

  // stable ascending rank of table values (invalid -> +inf), ties by index
  for (int i = tid; i < T; i += bd) {
    int vi = (sKeys[i] != PADK) ? sVals[i] : IMAX;
    int r = 0;
    for (int j = 0; j < T; ++j) {
      int vj = (sKeys[j] != PADK) ? sVals[j] : IMAX;
      r += (vj < vi) || ((vj == vi) && (j < i));
    }
    rnk[i] = r;
  }
  __syncthreads();

  int nRemove = nExist - T + nNew;
  if (nRemove < 0) nRemove = 0;

  // bump counts of kept keys present in this batch (binary search sorted uniques)
  for (int i = tid; i < T; i += bd) {
    int k = sKeys[i];
    bool keep = (k != PADK) && (rnk[i] >= nRemove);
    if (keep) {
      int lo = 0, hi = nValidU;
      while (lo < hi) { int mid = (lo + hi) >> 1; if (uq[mid] < k) lo = mid + 1; else hi = mid; }
      if (lo < nValidU && uq[lo] == k) sVals[i] += 1;
    }
  }

  // default PAD fill for outputs
  for (int i = tid; i < 2 * T; i += bd) {
    if (i < T) outK[i] = -1.0f; else outV[i - T] = -1.0f;
  }
  __syncthreads();

  // serial compaction (T ~ 2457 iterations, negligible)
  if (tid == 0) {
    int kpos = 0;
    for (int i = 0; i < T; ++i) {
      int k = sKeys[i];
      if (k != PADK && rnk[i] >= nRemove) {
        outK[kpos] = (float)k;
        outV[kpos] = (float)sVals[i];
        ++kpos;
      }
    }
    for (int u = 0; u < nValidU; ++u) {
      if (newm[u]) { outK[kpos] = (float)uq[u]; outV[kpos] = 1.0f; ++kpos; }
    }
  }
}

// ---------------- launcher ----------------
extern "C" void kernel_launch(void* const* d_in, const int* in_sizes, int n_in,
                              void* d_out, int out_size, void* d_ws, size_t ws_size,
                              hipStream_t stream) {
  const int* flat  = (const int*)d_in[0];   // inputs, N int32
  const int* vocab = (const int*)d_in[1];   // vocabulary, V int32
  const int* tkeys = (const int*)d_in[2];   // table_keys, T int32
  const int* tvals = (const int*)d_in[3];   // table_values, T int32
  const int N = in_sizes[0];
  const int V = in_sizes[1];
  const int T = in_sizes[2];
  float* out = (float*)d_out;               // [N lookup | T keys | T vals]

  int2* vslots = (int2*)d_ws;                                   // VHS * 8 B
  int*  ukeys  = (int*)((char*)d_ws + (size_t)VHS * sizeof(int2)); // UHS * 4 B

  const int initN = (VHS > UHS ? VHS : UHS);
  k_init<<<(initN + 255) / 256, 256, 0, stream>>>(vslots, ukeys);
  k_vocab_insert<<<(V + 255) / 256, 256, 0, stream>>>(vocab, V, vslots);
  k_lookup<<<256, 256, 0, stream>>>(flat, N, vslots, ukeys, out, V);
  k_finalize<<<1, 1024, 0, stream>>>(ukeys, tkeys, tvals, T, out + N, out + N + T);
}